// HashEncoding_19335942766824
// MI455X (gfx1250) — compile-verified
//
#include <hip/hip_runtime.h>
#include <cmath>

// ---------------------------------------------------------------------------
// Instant-NGP style multiresolution hash encoding (gather-only, no interp).
//   points: [N,3] f32 in [-1,1]; tables: [16, 2^19, 2] f32; out: [N, 32] f32.
// Memory-bound: 512MB output stream dominates; tables (64MB) live in L2.
// CDNA5 paths used: wave32 tiling, LDS staging (320KB/WGP),
// global_store_async_from_lds_b128 (ASYNCcnt) for coalesced 512B/wave stores,
// NT load hints for the streamed points. WMMA is inapplicable (zero matmul).
// ---------------------------------------------------------------------------

#define NUM_LEVELS   16
#define LOG2_HASH    19
#define HASHMAP_SIZE (1u << LOG2_HASH)
#define HASH_MASK    (HASHMAP_SIZE - 1u)
#define PRIME1       2654435761u
#define PRIME2       805459861u

#define BLOCK   256           // 8 waves (wave32)
#define ROW_DW  36            // 32 feature dwords + 4 pad dwords (144B, 16B aligned,
                              // 36 mod 64 gives only 2-way LDS bank conflicts on writes)

struct HEParams { float scale[NUM_LEVELS]; };   // (res_l - 1) as exact f32

// Native clang vector (NOT HIP's float4 class): required by the nontemporal
// builtins and usable on both host and device passes.
typedef float he_v4f __attribute__((vector_size(4 * sizeof(float))));

#if __has_builtin(__builtin_amdgcn_global_store_async_from_lds_b128)
#define HE_ASYNC_STORE 1
// Builtin signature (from the round-1 diagnostic): arg0 = int4 addrspace(1)*,
// arg1 = int4 addrspace(3)*, then imm offset, imm cpol.
typedef int he_v4i __attribute__((vector_size(4 * sizeof(int))));
typedef __attribute__((address_space(1))) he_v4i* he_gptr_t;
typedef __attribute__((address_space(3))) he_v4i* he_lptr_t;
#endif

__global__ __launch_bounds__(BLOCK) void he_main_kernel(
    const float* __restrict__ pts,
    const float* __restrict__ tbl,
    float* __restrict__ out,
    HEParams P)
{
    __shared__ __align__(16) float lds[BLOCK * ROW_DW];   // 36 KB

    const int tid = threadIdx.x;
    const size_t gp = (size_t)blockIdx.x * BLOCK + tid;   // global point id (full blocks only)

    // Streamed point read: NT hint keeps L2 for the hash tables.
    const float x = __builtin_nontemporal_load(pts + 3 * gp + 0);
    const float y = __builtin_nontemporal_load(pts + 3 * gp + 1);
    const float z = __builtin_nontemporal_load(pts + 3 * gp + 2);

    // Match jax op order exactly: clip((p+1)*0.5, 0, 1)
    const float px = fminf(fmaxf((x + 1.0f) * 0.5f, 0.0f), 1.0f);
    const float py = fminf(fmaxf((y + 1.0f) * 0.5f, 0.0f), 1.0f);
    const float pz = fminf(fmaxf((z + 1.0f) * 0.5f, 0.0f), 1.0f);

    // 16 independent gathers: fully unrolled -> 16 b64 loads in flight.
    float2 f[NUM_LEVELS];
#pragma unroll
    for (int l = 0; l < NUM_LEVELS; ++l) {
        const float s = P.scale[l];
        const unsigned cx = (unsigned)(px * s);   // trunc == floor for >= 0
        const unsigned cy = (unsigned)(py * s);
        const unsigned cz = (unsigned)(pz * s);
        const unsigned h = (cx ^ (cy * PRIME1) ^ (cz * PRIME2)) & HASH_MASK;
        f[l] = *((const float2*)tbl + (((size_t)l << LOG2_HASH) + h));
    }

    // Stage this point's 128B output row in LDS (padded stride).
    const int row = tid * ROW_DW;
#pragma unroll
    for (int l = 0; l < NUM_LEVELS; ++l)
        *(float2*)&lds[row + 2 * l] = f[l];

    // DS ops and ASYNC ops use different counters -> explicit ordering needed.
    asm volatile("s_wait_dscnt 0x0" ::: "memory");

    // Each wave drains its own 32 rows (written by its own lanes only, so no
    // cross-wave barrier needed) as 8 fully-coalesced 512B store instructions.
    const int wave = tid >> 5;
    const int lane = tid & 31;
    float* const wout = out + ((size_t)blockIdx.x * BLOCK + (size_t)wave * 32) * (2 * NUM_LEVELS);
#pragma unroll
    for (int i = 0; i < 8; ++i) {
        const int c = i * 32 + lane;   // chunk 0..255 of this wave's 4KB tile
        const int p = c >> 3;          // point within wave
        const int q = c & 7;           // 16B chunk within point row
        float* const gdst = wout + (size_t)c * 4;
        float* const lsrc = &lds[(wave * 32 + p) * ROW_DW + q * 4];
#ifdef HE_ASYNC_STORE
        __builtin_amdgcn_global_store_async_from_lds_b128(
            (he_gptr_t)gdst, (he_lptr_t)lsrc, 0, 0);
#else
        const he_v4f v = *(const he_v4f*)lsrc;
        __builtin_nontemporal_store(v, (he_v4f*)gdst);
#endif
    }
#ifdef HE_ASYNC_STORE
    asm volatile("s_wait_asynccnt 0x0" ::: "memory");
#endif
}

// Tail kernel: one thread per (point, level); unused when N % 256 == 0
// (N = 4194304 here), included for generality.
__global__ void he_tail_kernel(const float* __restrict__ pts,
                               const float* __restrict__ tbl,
                               float* __restrict__ out,
                               HEParams P, int start, int n)
{
    const int task = blockIdx.x * blockDim.x + threadIdx.x;
    const int p = start + (task >> 4);
    const int l = task & 15;
    if (p >= n) return;

    const float x = pts[3 * (size_t)p + 0];
    const float y = pts[3 * (size_t)p + 1];
    const float z = pts[3 * (size_t)p + 2];
    const float px = fminf(fmaxf((x + 1.0f) * 0.5f, 0.0f), 1.0f);
    const float py = fminf(fmaxf((y + 1.0f) * 0.5f, 0.0f), 1.0f);
    const float pz = fminf(fmaxf((z + 1.0f) * 0.5f, 0.0f), 1.0f);

    const float s = P.scale[l];
    const unsigned cx = (unsigned)(px * s);
    const unsigned cy = (unsigned)(py * s);
    const unsigned cz = (unsigned)(pz * s);
    const unsigned h = (cx ^ (cy * PRIME1) ^ (cz * PRIME2)) & HASH_MASK;
    const float2 f = *((const float2*)tbl + (((size_t)l << LOG2_HASH) + h));
    out[(size_t)p * 32 + 2 * l + 0] = f.x;
    out[(size_t)p * 32 + 2 * l + 1] = f.y;
}

extern "C" void kernel_launch(void* const* d_in, const int* in_sizes, int n_in,
                              void* d_out, int out_size, void* d_ws, size_t ws_size,
                              hipStream_t stream)
{
    (void)n_in; (void)out_size; (void)d_ws; (void)ws_size;

    const float* pts = (const float*)d_in[0];
    const float* tbl = (const float*)d_in[1];
    float* out = (float*)d_out;
    const int n = in_sizes[0] / 3;

    // Replicate the reference's resolution computation with the SAME double-
    // precision expression order as numpy:
    //   res_i = int(floor(16 * exp(i * log(2048/16) / 15)))
    // (level 15 lands on the 2048.0 knife edge; libm double math mirrors
    //  numpy's backend instead of a hand-rounded table).
    HEParams P;
    const double L = log(128.0);
    for (int i = 0; i < NUM_LEVELS; ++i) {
        const int res = (int)floor(16.0 * exp(((double)i * L) / 15.0));
        P.scale[i] = (float)(res - 1);   // exact in f32 (<= 2047)
    }

    const int nfull = n & ~(BLOCK - 1);
    if (nfull > 0) {
        he_main_kernel<<<nfull / BLOCK, BLOCK, 0, stream>>>(pts, tbl, out, P);
    }
    const int rem = n - nfull;
    if (rem > 0) {
        const int tasks = rem * NUM_LEVELS;
        he_tail_kernel<<<(tasks + 255) / 256, 256, 0, stream>>>(pts, tbl, out, P, nfull, n);
    }
}